// YoloV5_89850715832617
// MI455X (gfx1250) — compile-verified
//
#include <hip/hip_runtime.h>
#include <hip/hip_bf16.h>

#define N_CAND 25200
#define NP     25216          // padded stride (multiple of 32)
#define NCLS   80
#define B_IMG  16
#define MAXDET 300
#define CPB    256            // candidates per block, stage 1
#define TPB2   1024           // threads, stage 2
#define kNEG   (-1e30f)
#define CONF_T 0.25f
#define IOU_T  0.45f
#define MAX_WH 4096.0f

// ---------------- CDNA5 async global->LDS copy (ASYNCcnt path) --------------
#if __has_builtin(__builtin_amdgcn_global_load_async_to_lds_b128)
#define HAS_ASYNC_LDS 1
#else
#define HAS_ASYNC_LDS 0
#endif

__device__ __forceinline__ void async_copy_b128(const void* g, void* l) {
#if HAS_ASYNC_LDS
    typedef int v4i __attribute__((vector_size(16)));
    __builtin_amdgcn_global_load_async_to_lds_b128(
        (__attribute__((address_space(1))) v4i*)const_cast<void*>(g),
        (__attribute__((address_space(3))) v4i*)l,
        /*offset=*/0, /*cpol=*/0);
#else
    *(float4*)l = *(const float4*)g;
#endif
}

__device__ __forceinline__ void wait_async0() {
#if HAS_ASYNC_LDS
#if __has_builtin(__builtin_amdgcn_s_wait_asynccnt)
    __builtin_amdgcn_s_wait_asynccnt(0);
#else
    asm volatile("s_wait_asynccnt 0" ::: "memory");
#endif
#endif
}

// ---------------- Stage 1: decode candidates --------------------------------
// grid = (ceil(N/CPB), B).  Stages 256 rows (85 floats each) through LDS via
// async b128 DMA (coalesced), then each lane decodes one candidate.
__global__ __launch_bounds__(CPB) void yolo_prep(
    const float* __restrict__ pred,   // [B, N, 85]
    float*  __restrict__ score,       // [B, NP]
    float*  __restrict__ jcls,        // [B, NP]
    float4* __restrict__ nbox)        // [B, NP] (xyxy + class*MAX_WH)
{
    __shared__ float tile[CPB * 85];  // 87,040 B
    const int b  = blockIdx.y;
    const int n0 = blockIdx.x * CPB;
    const int cnt = min(CPB, N_CAND - n0);
    const float* src = pred + ((size_t)b * N_CAND + n0) * 85;

    // cnt*85 divisible by 4 for full (21760) and tail (9520) blocks; base 16B
    // aligned -> pure b128 DMA.
    const int nchunks = (cnt * 85) >> 2;
    for (int c = threadIdx.x; c < nchunks; c += CPB)
        async_copy_b128(src + c * 4, &tile[c * 4]);
    wait_async0();
    __syncthreads();

    const int t = threadIdx.x;
    if (t < cnt) {
        const float* p = &tile[t * 85];
        const float x = p[0], y = p[1], w = p[2], h = p[3], obj = p[4];
        float best = p[5]; int bj = 0;
        #pragma unroll 8
        for (int c = 1; c < NCLS; ++c) {
            const float v = p[5 + c];
            if (v > best) { best = v; bj = c; }   // first-occurrence argmax
        }
        const float conf  = best * obj;
        const bool  valid = (obj > CONF_T) && (conf > CONF_T);
        const float x1 = x - w * 0.5f, y1 = y - h * 0.5f;
        const float x2 = x + w * 0.5f, y2 = y + h * 0.5f;
        const float off = (float)bj * MAX_WH;
        const int gi = b * NP + n0 + t;
        score[gi] = valid ? conf : kNEG;
        jcls[gi]  = (float)bj;
        nbox[gi]  = make_float4(x1 + off, y1 + off, x2 + off, y2 + off);
    }
}

// ---------------- Stage 2: serial NMS, scores LDS-resident ------------------
// One block (32 wave32 waves) per image. Live-score vector (100.8 KB) lives in
// LDS for the whole 300-step scan; boxes stream from L2 (6.4 MB total, fully
// L2-resident).  Each step does ONE fused pass: suppress-by-previous-box, then
// argmax, with the global box load gated on liveness.
__global__ __launch_bounds__(TPB2) void yolo_nms(
    const float*  __restrict__ score_g,
    const float*  __restrict__ jcls,
    const float4* __restrict__ nbox,
    float* __restrict__ out_rows,     // [B, 300, 6]
    float* __restrict__ out_keep)     // [B, 300]
{
    __shared__ float  s_score[NP];    // 100,864 B
    __shared__ float  s_rv[32];
    __shared__ int    s_ri[32];
    __shared__ float4 s_box;
    __shared__ int    s_done;

    const int b   = blockIdx.x;
    const int tid = threadIdx.x;
    const float*  sg = score_g + b * NP;
    const float4* nb = nbox    + b * NP;
    const float*  jc = jcls    + b * NP;
    float* rows  = out_rows + b * MAXDET * 6;
    float* keepv = out_keep + b * MAXDET;

    // DMA the live-score vector into LDS once (6300 x b128).
    for (int c = tid; c < (N_CAND >> 2); c += TPB2)
        async_copy_b128(sg + c * 4, &s_score[c * 4]);
    if (tid < NP - N_CAND) s_score[N_CAND + tid] = kNEG;
    if (tid == 0) {
        s_done = 0;
        // Sentinel: IoU with any real box (coords >= -60) is exactly 0.
        s_box  = make_float4(-1.0e8f, -1.0e8f, -9.9e7f, -9.9e7f);
    }
    wait_async0();
    __syncthreads();

    int t;
    for (t = 0; t < MAXDET; ++t) {
        // ---- fused pass: kill by previous selection, then argmax ----
        const float4 B4 = s_box;
        const float a1 = (B4.z - B4.x) * (B4.w - B4.y);
        float bv = -__builtin_inff();
        int   bi = 0x7fffffff;
        for (int i = tid; i < N_CAND; i += TPB2) {
            float v = s_score[i];
            if (v > 0.0f) {                 // live (valid scores are > 0.25)
                const float4 c = nb[i];
                const float xx1 = fmaxf(B4.x, c.x), yy1 = fmaxf(B4.y, c.y);
                const float xx2 = fminf(B4.z, c.z), yy2 = fminf(B4.w, c.w);
                const float inter = fmaxf(xx2 - xx1, 0.0f) * fmaxf(yy2 - yy1, 0.0f);
                const float a2 = (c.z - c.x) * (c.w - c.y);
                if (inter > IOU_T * (a1 + a2 - inter)) {   // iou > IOU_T (denom > 0)
                    v = kNEG;
                    s_score[i] = kNEG;
                }
            }
            if (v > bv || (v == bv && i < bi)) { bv = v; bi = i; }
        }
        // ---- wave then block reduction (first-occurrence tie break) ----
        #pragma unroll
        for (int o = 16; o > 0; o >>= 1) {
            const float ov = __shfl_down(bv, o, 32);
            const int   oi = __shfl_down(bi, o, 32);
            if (ov > bv || (ov == bv && oi < bi)) { bv = ov; bi = oi; }
        }
        const int wave = tid >> 5, lane = tid & 31;
        if (lane == 0) { s_rv[wave] = bv; s_ri[wave] = bi; }
        __syncthreads();
        if (wave == 0) {
            bv = s_rv[lane]; bi = s_ri[lane];
            #pragma unroll
            for (int o = 16; o > 0; o >>= 1) {
                const float ov = __shfl_down(bv, o, 32);
                const int   oi = __shfl_down(bi, o, 32);
                if (ov > bv || (ov == bv && oi < bi)) { bv = ov; bi = oi; }
            }
            if (lane == 0) {
                const float sel = bv;
                if (sel > 0.0f) {
                    const float4 box = nb[bi];
                    const float  j   = jc[bi];
                    s_box = box;
                    s_score[bi] = kNEG;
                    const float o2 = j * MAX_WH;
                    rows[t * 6 + 0] = box.x - o2;
                    rows[t * 6 + 1] = box.y - o2;
                    rows[t * 6 + 2] = box.z - o2;
                    rows[t * 6 + 3] = box.w - o2;
                    rows[t * 6 + 4] = sel;
                    rows[t * 6 + 5] = j;
                    keepv[t] = 1.0f;
                } else {
                    s_done = 1;
                }
            }
        }
        __syncthreads();
        if (s_done) break;
    }

    // zero the tail (reference multiplies rows by keeps -> zeros)
    for (int k = t * 6 + tid; k < MAXDET * 6; k += TPB2) rows[k]  = 0.0f;
    for (int k = t     + tid; k < MAXDET;     k += TPB2) keepv[k] = 0.0f;
}

// ---------------- host launch ----------------------------------------------
extern "C" void kernel_launch(void* const* d_in, const int* in_sizes, int n_in,
                              void* d_out, int out_size, void* d_ws, size_t ws_size,
                              hipStream_t stream) {
    const float* pred = (const float*)d_in[0];

    float*  ws    = (float*)d_ws;
    float*  score = ws;                                  // B*NP
    float*  jcl   = ws +     (size_t)B_IMG * NP;         // B*NP
    float4* nbx   = (float4*)(ws + 2 * (size_t)B_IMG * NP); // B*NP float4 (16B aligned)

    float* rows = (float*)d_out;                         // [16,300,6]
    float* keep = rows + (size_t)B_IMG * MAXDET * 6;     // [16,300]

    dim3 g1((N_CAND + CPB - 1) / CPB, B_IMG);
    yolo_prep<<<g1, CPB, 0, stream>>>(pred, score, jcl, nbx);
    yolo_nms<<<B_IMG, TPB2, 0, stream>>>(score, jcl, nbx, rows, keep);
}